// AKT_39728447488478
// MI455X (gfx1250) — compile-verified
//
#include <hip/hip_runtime.h>
#include <hip/hip_bf16.h>

#define BATCH 32
#define SEQ   512
#define DIM   512
#define HEADS 8
#define DKH   64
#define DFFN  2048
#define MROWS (BATCH * SEQ)   // 16384
#define SATT  516             // padded LDS row stride (floats)

typedef float v2f __attribute__((ext_vector_type(2)));
typedef float v8f __attribute__((ext_vector_type(8)));

__device__ __forceinline__ v8f wmma4(v2f a, v2f b, v8f c) {
    // V_WMMA_F32_16X16X4_F32 : D = A(16x4 f32) * B(4x16 f32) + C(16x16 f32)
    return __builtin_amdgcn_wmma_f32_16x16x4_f32(
        false, a, false, b, (short)0, c, false, false);
}

// ---------------------------------------------------------------------------
// Generic GEMM: C[M,N] = A[M,K] @ W[N,K]^T + bias[N] (optional relu)
// 256 threads = 8 waves; each wave owns one 16x16 C tile.
// ---------------------------------------------------------------------------
__global__ void gemm_bias(const float* __restrict__ A, const float* __restrict__ W,
                          const float* __restrict__ bias, float* __restrict__ C,
                          int M, int N, int K, int relu) {
    int wave = threadIdx.x >> 5;
    int lane = threadIdx.x & 31;
    int half = lane >> 4;
    int r    = lane & 15;
    long ntiles = N >> 4;
    long t  = (long)blockIdx.x * 8 + wave;
    int  mt = (int)(t / ntiles) * 16;
    int  nt = (int)(t % ntiles) * 16;
    if (mt >= M) return;   // wave-uniform

    const float* arow = A + (size_t)(mt + r) * K + 4 * half;
    const float* brow = W + (size_t)(nt + r) * K + 4 * half;

    v8f acc = {};
    for (int k0 = 0; k0 < K; k0 += 8) {
        float4 a4 = *(const float4*)(arow + k0);
        float4 b4 = *(const float4*)(brow + k0);
        v2f a0 = {a4.x, a4.y}, a1 = {a4.z, a4.w};
        v2f b0 = {b4.x, b4.y}, b1 = {b4.z, b4.w};
        acc = wmma4(a0, b0, acc);
        acc = wmma4(a1, b1, acc);
    }
    float bn = bias[nt + r];
    for (int i = 0; i < 8; ++i) {
        int   m = mt + i + 8 * half;
        float v = acc[i] + bn;
        if (relu) v = fmaxf(v, 0.f);
        C[(size_t)m * N + nt + r] = v;
    }
}

// ---------------------------------------------------------------------------
// Fused attention for one (batch, head, 16-row slab).
// KQ buffer serves both q and k (kq_same in reference).
// maskIncl: 1 -> j<=i allowed ; 0 -> j<i allowed. zeroPad zeroes query row 0.
// ---------------------------------------------------------------------------
__global__ void attn_kernel(const float* __restrict__ KQ, const float* __restrict__ V,
                            float* __restrict__ O, const float* __restrict__ gam,
                            int maskIncl, int zeroPad) {
    __shared__ float sc[16][SATT];

    int blk  = blockIdx.x;
    int rt   = blk & 31;          // S/16 = 32 row slabs
    int h    = (blk >> 5) & 7;
    int b    = blk >> 8;
    int row0 = rt * 16;

    int wave = threadIdx.x >> 5;
    int lane = threadIdx.x & 31;
    int half = lane >> 4;
    int r    = lane & 15;
    size_t bs = (size_t)b * SEQ;

    // ---- Phase A: scores = (Q @ K^T) / sqrt(DK) into LDS -----------------
    const float* qbase = KQ + (bs + row0 + r) * DIM + h * DKH + 4 * half;
    for (int t = 0; t < 4; ++t) {
        int ct = wave * 4 + t;    // key column tile 0..31
        const float* kbase = KQ + (bs + ct * 16 + r) * DIM + h * DKH + 4 * half;
        v8f acc = {};
        for (int k0 = 0; k0 < DKH; k0 += 8) {
            float4 a4 = *(const float4*)(qbase + k0);
            float4 b4 = *(const float4*)(kbase + k0);
            v2f a0 = {a4.x, a4.y}, a1 = {a4.z, a4.w};
            v2f b0 = {b4.x, b4.y}, b1 = {b4.z, b4.w};
            acc = wmma4(a0, b0, acc);
            acc = wmma4(a1, b1, acc);
        }
        for (int i = 0; i < 8; ++i)
            sc[i + 8 * half][ct * 16 + r] = acc[i] * 0.125f; // 1/sqrt(64)
    }
    __syncthreads();

    // ---- Phase B: per-row softmax / distance decay / second softmax ------
    if (threadIdx.x < 16) {
        int rr = threadIdx.x;
        int ig = row0 + rr;
        float gamma = gam[h];
        float g = -(fmaxf(gamma, 0.f) + log1pf(expf(-fabsf(gamma)))); // -softplus
        const float NEGV = -1e32f;
        int lim = maskIncl ? ig : (ig - 1);   // allowed: j <= lim

        float m1 = -3.0e38f;
        for (int j = 0; j < SEQ; ++j) {
            float s = (j <= lim) ? sc[rr][j] : NEGV;
            m1 = fmaxf(m1, s);
        }
        float sumAll = 0.f, sumM = 0.f;
        for (int j = 0; j < SEQ; ++j) {
            float s = (j <= lim) ? sc[rr][j] : NEGV;
            float e = expf(s - m1);
            sumAll += e;
            if (j <= lim) sumM += e;
        }
        float invAll = 1.f / sumAll;
        float cum = 0.f, m2 = -3.0e38f;
        for (int j = 0; j < SEQ; ++j) {
            float s2;
            if (j <= lim) {
                float s = sc[rr][j];
                float e = expf(s - m1);
                cum += e;
                float remain = (sumM - cum) * invAll;        // disttot - distcum
                float pos = fabsf((float)(j - ig));
                float dv  = sqrtf(fmaxf(remain * pos, 0.f));
                float te  = expf(dv * g);
                te = fminf(fmaxf(te, 1e-5f), 1e5f);
                s2 = s * te;
            } else {
                s2 = NEGV;
            }
            sc[rr][j] = s2;
            m2 = fmaxf(m2, s2);
        }
        float sum2 = 0.f;
        for (int j = 0; j < SEQ; ++j) {
            float e = expf(sc[rr][j] - m2);
            sc[rr][j] = e;
            sum2 += e;
        }
        float scale = 1.f / sum2;
        if (zeroPad && ig == 0) scale = 0.f;
        for (int j = 0; j < SEQ; ++j) sc[rr][j] *= scale;
    }
    __syncthreads();

    // ---- Phase C: out = attn @ V, stored to [B,S,D] ----------------------
    if (wave < 4) {                 // wave-uniform; DK/16 = 4 column tiles
        int nt = wave * 16;
        const float* vb = V + bs * DIM + h * DKH + nt + r;
        v8f acc = {};
        for (int k0 = 0; k0 < SEQ; k0 += 8) {
            int kk = k0 + 4 * half;
            float4 a4 = *(const float4*)&sc[r][kk];
            v2f a0 = {a4.x, a4.y}, a1 = {a4.z, a4.w};
            const float* vp = vb + (size_t)kk * DIM;
            v2f b0 = {vp[0],        vp[DIM]};
            v2f b1 = {vp[2 * DIM],  vp[3 * DIM]};
            acc = wmma4(a0, b0, acc);
            acc = wmma4(a1, b1, acc);
        }
        for (int i = 0; i < 8; ++i)
            O[(bs + row0 + i + 8 * half) * DIM + h * DKH + nt + r] = acc[i];
    }
}

// ---------------------------------------------------------------------------
// Residual add + LayerNorm, in place on x. One block (256 thr) per row, D=512.
// ---------------------------------------------------------------------------
__global__ void resid_ln(float* __restrict__ x, const float* __restrict__ dx,
                         const float* __restrict__ g, const float* __restrict__ b) {
    __shared__ float red[256];
    int tid = threadIdx.x;
    size_t base = (size_t)blockIdx.x * DIM;
    float v0 = x[base + tid]       + dx[base + tid];
    float v1 = x[base + 256 + tid] + dx[base + 256 + tid];

    red[tid] = v0 + v1;
    __syncthreads();
    for (int off = 128; off > 0; off >>= 1) {
        if (tid < off) red[tid] += red[tid + off];
        __syncthreads();
    }
    float mu = red[0] * (1.f / 512.f);
    __syncthreads();

    float d0 = v0 - mu, d1 = v1 - mu;
    red[tid] = d0 * d0 + d1 * d1;
    __syncthreads();
    for (int off = 128; off > 0; off >>= 1) {
        if (tid < off) red[tid] += red[tid + off];
        __syncthreads();
    }
    float rstd = rsqrtf(red[0] * (1.f / 512.f) + 1e-5f);

    x[base + tid]       = d0 * rstd * g[tid]       + b[tid];
    x[base + 256 + tid] = d1 * rstd * g[256 + tid] + b[256 + tid];
}

// ---------------------------------------------------------------------------
// Fused FFN: Out = relu(X @ W1^T + b1) @ W2^T + b2 (no 16384x2048 spill;
// hidden chunked 512 wide through LDS, W2 accumulation lives in registers).
// One block (8 waves) per 16-row slab.
// ---------------------------------------------------------------------------
__global__ void ffn_fused(const float* __restrict__ X,  const float* __restrict__ W1,
                          const float* __restrict__ b1, const float* __restrict__ W2,
                          const float* __restrict__ b2, float* __restrict__ Out) {
    __shared__ float hbuf[16][SATT];
    int mt   = blockIdx.x * 16;
    int wave = threadIdx.x >> 5;
    int lane = threadIdx.x & 31;
    int half = lane >> 4;
    int r    = lane & 15;

    const float* xb = X + (size_t)(mt + r) * DIM + 4 * half;
    v8f acc2[4] = {};

    for (int c = 0; c < 4; ++c) {          // DFF chunks of 512
        int f0 = c * 512;
        // phase 1: h[:, f0:f0+512] = relu(X @ W1^T + b1)
        for (int t = 0; t < 4; ++t) {
            int ft = (wave * 4 + t) * 16;
            const float* w1b = W1 + (size_t)(f0 + ft + r) * DIM + 4 * half;
            v8f acc = {};
            for (int k0 = 0; k0 < DIM; k0 += 8) {
                float4 a4 = *(const float4*)(xb + k0);
                float4 b4 = *(const float4*)(w1b + k0);
                v2f a0 = {a4.x, a4.y}, a1 = {a4.z, a4.w};
                v2f b0 = {b4.x, b4.y}, b1v = {b4.z, b4.w};
                acc = wmma4(a0, b0, acc);
                acc = wmma4(a1, b1v, acc);
            }
            float bb = b1[f0 + ft + r];
            for (int i = 0; i < 8; ++i)
                hbuf[i + 8 * half][ft + r] = fmaxf(acc[i] + bb, 0.f);
        }
        __syncthreads();
        // phase 2: accumulate Out += h_chunk @ W2_chunk^T
        for (int t = 0; t < 4; ++t) {
            int nt = (wave * 4 + t) * 16;
            const float* w2b = W2 + (size_t)(nt + r) * DFFN + f0 + 4 * half;
            v8f acc = acc2[t];
            for (int k0 = 0; k0 < 512; k0 += 8) {
                float4 a4 = *(const float4*)&hbuf[r][k0 + 4 * half];
                float4 b4 = *(const float4*)(w2b + k0);
                v2f a0 = {a4.x, a4.y}, a1 = {a4.z, a4.w};
                v2f b0 = {b4.x, b4.y}, b1v = {b4.z, b4.w};
                acc = wmma4(a0, b0, acc);
                acc = wmma4(a1, b1v, acc);
            }
            acc2[t] = acc;
        }
        __syncthreads();
    }
    for (int t = 0; t < 4; ++t) {
        int nt = (wave * 4 + t) * 16;
        float bb = b2[nt + r];
        for (int i = 0; i < 8; ++i)
            Out[(size_t)(mt + i + 8 * half) * DIM + nt + r] = acc2[t][i] + bb;
    }
}

// ---------------------------------------------------------------------------
extern "C" void kernel_launch(void* const* d_in, const int* in_sizes, int n_in,
                              void* d_out, int out_size, void* d_ws, size_t ws_size,
                              hipStream_t stream) {
    (void)in_sizes; (void)n_in; (void)out_size; (void)ws_size;

    const float* q_embed  = (const float*)d_in[0];
    const float* qa_embed = (const float*)d_in[1];
    const float* Wk   = (const float*)d_in[2];
    const float* bk   = (const float*)d_in[3];
    const float* Wv   = (const float*)d_in[4];
    const float* bv   = (const float*)d_in[5];
    const float* Wo   = (const float*)d_in[6];
    const float* bo   = (const float*)d_in[7];
    const float* gam  = (const float*)d_in[8];
    const float* ln1g = (const float*)d_in[9];
    const float* ln1b = (const float*)d_in[10];
    const float* W1   = (const float*)d_in[11];
    const float* b1   = (const float*)d_in[12];
    const float* W2   = (const float*)d_in[13];
    const float* b2   = (const float*)d_in[14];
    const float* ln2g = (const float*)d_in[15];
    const float* ln2b = (const float*)d_in[16];

    const size_t NELEM = (size_t)MROWS * DIM;        // 8,388,608 floats
    float* xbuf = (float*)d_ws;
    float* kq   = xbuf + NELEM;
    float* vv   = kq + NELEM;
    float* tmp  = vv + NELEM;

    hipMemcpyAsync(xbuf, q_embed, NELEM * sizeof(float),
                   hipMemcpyDeviceToDevice, stream);

    const int gemmBlocks = (MROWS / 16) * (DIM / 16) / 8;   // 4096
    const int attnBlocks = BATCH * HEADS * (SEQ / 16);      // 8192

    for (int l = 0; l < 4; ++l) {
        int first = (l % 2 == 0);
        const float* Wk_l = Wk + (size_t)l * DIM * DIM;
        const float* Wv_l = Wv + (size_t)l * DIM * DIM;
        const float* Wo_l = Wo + (size_t)l * DIM * DIM;
        const float* W1_l = W1 + (size_t)l * DFFN * DIM;
        const float* W2_l = W2 + (size_t)l * DIM * DFFN;

        // q/k projection (kq_same): kq = x @ Wk^T + bk
        gemm_bias<<<gemmBlocks, 256, 0, stream>>>(
            xbuf, Wk_l, bk + l * DIM, kq, MROWS, DIM, DIM, 0);
        // v projection: vv = vals @ Wv^T + bv
        const float* vals = first ? qa_embed : xbuf;
        gemm_bias<<<gemmBlocks, 256, 0, stream>>>(
            vals, Wv_l, bv + l * DIM, vv, MROWS, DIM, DIM, 0);
        // fused attention -> tmp
        attn_kernel<<<attnBlocks, 256, 0, stream>>>(
            kq, vv, tmp, gam + l * HEADS, first ? 0 : 1, first ? 1 : 0);
        // output projection -> kq (reuse)
        gemm_bias<<<gemmBlocks, 256, 0, stream>>>(
            tmp, Wo_l, bo + l * DIM, kq, MROWS, DIM, DIM, 0);
        // x = LN(x + proj)
        resid_ln<<<MROWS, 256, 0, stream>>>(
            xbuf, kq, ln1g + l * DIM, ln1b + l * DIM);

        if (first) {
            ffn_fused<<<MROWS / 16, 256, 0, stream>>>(
                xbuf, W1_l, b1 + l * DFFN, W2_l, b2 + l * DIM, kq);
            resid_ln<<<MROWS, 256, 0, stream>>>(
                xbuf, kq, ln2g + l * DIM, ln2b + l * DIM);
        }
    }

    hipMemcpyAsync(d_out, xbuf, NELEM * sizeof(float),
                   hipMemcpyDeviceToDevice, stream);
}